// Similarity_Loss_1_vs_all_22256520528839
// MI455X (gfx1250) — compile-verified
//
#include <hip/hip_runtime.h>
#include <hip/hip_bf16.h>
#include <math.h>

#define NB        4096
#define ND        1024
#define INV_TEMP  20.0f          // 1 / 0.05
#define MSENT     (-1.0e30f)     // finite "-inf" sentinel: no NaN paths

typedef __attribute__((ext_vector_type(2))) float v2f;
typedef __attribute__((ext_vector_type(8))) float v8f;

#if defined(__has_builtin)
#  if __has_builtin(__builtin_amdgcn_wmma_f32_16x16x4_f32)
#    define HAVE_WMMA_F32 1
#  endif
#  if __has_builtin(__builtin_amdgcn_sqrtf)
#    define FAST_SQRT(x) __builtin_amdgcn_sqrtf(x)
#  endif
#endif
#ifndef FAST_SQRT
#  define FAST_SQRT(x) sqrtf(x)
#endif

// branchless logsumexp merge (safe with MSENT sentinel: exp underflows to 0)
__device__ __forceinline__ void lse_combine(float& m1, float& s1, float m2, float s2) {
    float mm = fmaxf(m1, m2);
    s1 = s1 * __expf(m1 - mm) + s2 * __expf(m2 - mm);
    m1 = mm;
}

// ---------------------------------------------------------------------------
// Kernel 1: squared row norms of T (blocks 0..NB-1) and M (blocks NB..2NB-1)
// ---------------------------------------------------------------------------
__global__ void row_norms_kernel(const float* __restrict__ Tm,
                                 const float* __restrict__ Mm,
                                 float* __restrict__ nt,
                                 float* __restrict__ nm) {
    const int b   = blockIdx.x;              // 0 .. 2*NB-1
    const int row = b & (NB - 1);
    const float* src = (b < NB) ? Tm : Mm;

    float acc = 0.0f;
    for (int k = threadIdx.x; k < ND; k += blockDim.x) {
        float v = src[(size_t)row * ND + k];
        acc += v * v;
    }
    #pragma unroll
    for (int off = 16; off > 0; off >>= 1)
        acc += __shfl_down(acc, off, 32);

    __shared__ float red[8];
    const int lane = threadIdx.x & 31;
    const int wave = threadIdx.x >> 5;
    if (lane == 0) red[wave] = acc;
    __syncthreads();
    if (threadIdx.x == 0) {
        float t = 0.0f;
        const int nwaves = blockDim.x >> 5;
        for (int w = 0; w < nwaves; ++w) t += red[w];
        ((b < NB) ? nt : nm)[row] = t;
    }
}

// ---------------------------------------------------------------------------
// Kernel 2: fused distance-GEMM + streaming logsumexp.
// Grid: 256 blocks (16 rows each), 256 threads (8 waves).
// Each wave computes a 16x64 output strip per iteration: one A fragment is
// reused across 4 B tiles / 4 C accumulators (5 b64 loads per 4 WMMA).
// Diagonal is NOT masked in the hot loop; it is recorded and subtracted from
// the exp-sum at the end (identical math to the -inf mask).
// ---------------------------------------------------------------------------
__global__ void fused_lse_kernel(const float* __restrict__ Tm,
                                 const float* __restrict__ Mm,
                                 const float* __restrict__ nt,
                                 const float* __restrict__ nm,
                                 float* __restrict__ rowvals) {
    const int lane  = threadIdx.x & 31;
    const int wave  = threadIdx.x >> 5;      // 0..7
    const int half  = lane >> 4;             // 0 or 1 (C rows 0-7 vs 8-15)
    const int l16   = lane & 15;
    const int row_base = blockIdx.x * 16;
    const int arow  = row_base + l16;        // A-matrix row this lane carries

    __shared__ float wm[8][16];
    __shared__ float ws[8][16];
    __shared__ float diag[16];
    if (threadIdx.x < 16) diag[threadIdx.x] = 0.0f;
    __syncthreads();

    // per-lane streaming LSE state: element r -> global row row_base+8*half+r
    float m[8], s[8], ntv[8];
    #pragma unroll
    for (int r = 0; r < 8; ++r) {
        m[r]   = MSENT;
        s[r]   = 0.0f;
        ntv[r] = nt[row_base + 8 * half + r];
    }

    const float* aptr = Tm + (size_t)arow * ND + 2 * half;

    // 64 column groups of 64 cols; this wave takes cg = wave, wave+8, ...
    for (int cg = wave; cg < NB / 64; cg += 8) {
        const int col0  = cg * 64;
        const int bcol0 = col0 + l16;               // lane's col in tile 0

        // prefetch this wave's next column group's B rows
        if (bcol0 + 512 < NB)
            __builtin_prefetch(Mm + (size_t)(bcol0 + 512) * ND, 0, 1);

        const float* bp0 = Mm + (size_t)(bcol0     ) * ND + 2 * half;
        const float* bp1 = Mm + (size_t)(bcol0 + 16) * ND + 2 * half;
        const float* bp2 = Mm + (size_t)(bcol0 + 32) * ND + 2 * half;
        const float* bp3 = Mm + (size_t)(bcol0 + 48) * ND + 2 * half;

        v8f c[4] = {v8f{}, v8f{}, v8f{}, v8f{}};
#ifdef HAVE_WMMA_F32
        #pragma unroll 4
        for (int k = 0; k < ND; k += 4) {
            v2f a  = *(const v2f*)(aptr + k);
            v2f b0 = *(const v2f*)(bp0 + k);
            v2f b1 = *(const v2f*)(bp1 + k);
            v2f b2 = *(const v2f*)(bp2 + k);
            v2f b3 = *(const v2f*)(bp3 + k);
            c[0] = __builtin_amdgcn_wmma_f32_16x16x4_f32(false, a, false, b0, (short)0, c[0], false, false);
            c[1] = __builtin_amdgcn_wmma_f32_16x16x4_f32(false, a, false, b1, (short)0, c[1], false, false);
            c[2] = __builtin_amdgcn_wmma_f32_16x16x4_f32(false, a, false, b2, (short)0, c[2], false, false);
            c[3] = __builtin_amdgcn_wmma_f32_16x16x4_f32(false, a, false, b3, (short)0, c[3], false, false);
        }
#else
        // scalar fallback (keeps compile green; histogram would show wmma=0)
        #pragma unroll
        for (int t = 0; t < 4; ++t)
            #pragma unroll
            for (int r = 0; r < 8; ++r) {
                const int rr = row_base + 8 * half + r;
                float acc = 0.0f;
                for (int k = 0; k < ND; ++k)
                    acc = fmaf(Tm[(size_t)rr * ND + k],
                               Mm[(size_t)(bcol0 + 16 * t) * ND + k], acc);
                c[t][r] = acc;
            }
#endif
        // epilogue: logits + branchless streaming LSE (full EXEC throughout)
        #pragma unroll
        for (int t = 0; t < 4; ++t) {
            const int col = bcol0 + 16 * t;          // lane's C column for tile t
            const float nmv = nm[col];
            #pragma unroll
            for (int r = 0; r < 8; ++r) {
                const int rr = row_base + 8 * half + r;
                float sq   = ntv[r] + nmv - 2.0f * c[t][r];
                float x    = -INV_TEMP * FAST_SQRT(fmaxf(sq, 0.0f));
                if (col == rr) diag[8 * half + r] = x;   // unique writer per row
                float nm_  = fmaxf(m[r], x);
                s[r] = s[r] * __expf(m[r] - nm_) + __expf(x - nm_);
                m[r] = nm_;
            }
        }
    }

    // reduce across the 16 lanes sharing each C row (xor 1,2,4,8 stays in-half)
    #pragma unroll
    for (int off = 1; off < 16; off <<= 1) {
        #pragma unroll
        for (int r = 0; r < 8; ++r) {
            float mo = __shfl_xor(m[r], off, 32);
            float so = __shfl_xor(s[r], off, 32);
            lse_combine(m[r], s[r], mo, so);
        }
    }
    if (l16 == 0) {
        #pragma unroll
        for (int r = 0; r < 8; ++r) {
            wm[wave][8 * half + r] = m[r];
            ws[wave][8 * half + r] = s[r];
        }
    }
    __syncthreads();

    // combine 8 wave-partials per row; remove diag term from the sum, finish.
    if (threadIdx.x < 16) {
        const int t = threadIdx.x;
        float M0 = wm[0][t], S0 = ws[0][t];
        #pragma unroll
        for (int w = 1; w < 8; ++w)
            lse_combine(M0, S0, wm[w][t], ws[w][t]);
        float d  = diag[t];
        float se = S0 - __expf(d - M0);          // exclude self-term
        float lse = M0 + __logf(se);
        rowvals[row_base + t] = lse - d;
    }
}

// ---------------------------------------------------------------------------
// Kernel 3: mean of the 4096 per-row values -> scalar
// ---------------------------------------------------------------------------
__global__ void mean_kernel(const float* __restrict__ rowvals, float* __restrict__ out) {
    float acc = 0.0f;
    for (int i = threadIdx.x; i < NB; i += blockDim.x)
        acc += rowvals[i];
    #pragma unroll
    for (int off = 16; off > 0; off >>= 1)
        acc += __shfl_down(acc, off, 32);

    __shared__ float red[8];
    const int lane = threadIdx.x & 31;
    const int wave = threadIdx.x >> 5;
    if (lane == 0) red[wave] = acc;
    __syncthreads();
    if (threadIdx.x == 0) {
        float t = 0.0f;
        const int nwaves = blockDim.x >> 5;
        for (int w = 0; w < nwaves; ++w) t += red[w];
        out[0] = t * (1.0f / (float)NB);
    }
}

extern "C" void kernel_launch(void* const* d_in, const int* in_sizes, int n_in,
                              void* d_out, int out_size, void* d_ws, size_t ws_size,
                              hipStream_t stream) {
    const float* Tm = (const float*)d_in[0];   // text_embeddings  [4096,1024] f32
    const float* Mm = (const float*)d_in[1];   // image_embeddings [4096,1024] f32
    (void)in_sizes; (void)n_in; (void)out_size; (void)ws_size;

    float* ws      = (float*)d_ws;
    float* nt      = ws;               // 4096 floats
    float* nm      = ws + NB;          // 4096 floats
    float* rowvals = ws + 2 * NB;      // 4096 floats

    row_norms_kernel<<<2 * NB, 256, 0, stream>>>(Tm, Mm, nt, nm);
    fused_lse_kernel<<<NB / 16, 256, 0, stream>>>(Tm, Mm, nt, nm, rowvals);
    mean_kernel<<<1, 256, 0, stream>>>(rowvals, (float*)d_out);
}